// RaycastFeatures_42597485641917
// MI455X (gfx1250) — compile-verified
//
#include <hip/hip_runtime.h>
#include <stdint.h>

#define IGNORE_LABEL (-100)

typedef float f4  __attribute__((ext_vector_type(4)));
typedef int   v4i __attribute__((ext_vector_type(4)));

static constexpr int D = 128;               // feature dim (reference)
static constexpr int ROW_BYTES = D * 4;     // 512B per feature row
static constexpr int NSLOT = 4;             // pixels per wave per batch (2KB in flight)
static constexpr int WAVES_PER_BLOCK = 8;   // 256 threads, wave32

// ---- gfx1250 async global<->LDS path (guarded; falls back cleanly) --------
#if defined(__has_builtin)
#if __has_builtin(__builtin_amdgcn_global_load_async_to_lds_b128) && \
    __has_builtin(__builtin_amdgcn_global_store_async_from_lds_b128)
#define HAVE_GFX1250_ASYNC 1
#endif
#endif

// The builtins take typed vector pointers: (global v4i*, shared v4i*, int, int)
typedef __attribute__((address_space(1))) v4i gv4i;
typedef __attribute__((address_space(3))) v4i lv4i;

static __device__ __forceinline__ gv4i* as_gv4(const void* p) {
    return (gv4i*)(uintptr_t)p;
}
static __device__ __forceinline__ lv4i* as_lv4(const void* p) {
    // low 32 bits of a generic pointer into __shared__ are the DS byte address
    return (lv4i*)(uint32_t)(uintptr_t)p;
}
static __device__ __forceinline__ void wait_async0() {
#if defined(__has_builtin) && __has_builtin(__builtin_amdgcn_s_wait_asynccnt)
    __builtin_amdgcn_s_wait_asynccnt(0);
#else
    asm volatile("s_wait_asynccnt 0x0" ::: "memory");
#endif
}

// ---------------------------------------------------------------------------
// Kernel 0: zero histogram bins (d_out is poisoned; runs every call).
// ---------------------------------------------------------------------------
__global__ __launch_bounds__(256) void RaycastZeroCounts_kernel(int* __restrict__ counts,
                                                                int V) {
    int i = blockIdx.x * blockDim.x + threadIdx.x;
    if (i < V) counts[i] = 0;
}

#if defined(HAVE_GFX1250_ASYNC)
// ---------------------------------------------------------------------------
// Gather via the CDNA5 async copy engine: feature row -> LDS -> output, no
// VGPR round-trip for the 314MB bulk stream. One wave32 per NSLOT pixels;
// each lane moves 16B (b128) so a wave moves exactly one 512B row per op.
//   phase A: issue NSLOT async gather loads (global -> LDS slots)
//   s_wait_asynccnt 0
//   phase B: async store each row out (LDS -> global, TH=NT via cpol=1);
//            misses store zeros straight from VGPRs (never touch LDS).
// ---------------------------------------------------------------------------
__global__ __launch_bounds__(256) void RaycastGatherAsync_kernel(
    const float* __restrict__ feats,   // [V, 128]
    const int*   __restrict__ idx_img, // [P]
    float*       __restrict__ proj,    // [P, 128]
    float*       __restrict__ idx_out, // [P]
    int*         __restrict__ counts,  // [V]
    int P) {
    __shared__ alignas(16) unsigned char smem[WAVES_PER_BLOCK * NSLOT * ROW_BYTES];

    const int lane = threadIdx.x & 31;
    const int wib  = threadIdx.x >> 5;                       // wave in block
    const int wave = blockIdx.x * WAVES_PER_BLOCK + wib;
    const int pix0 = wave * NSLOT;
    if (pix0 >= P) return;

    unsigned char* slotbase = smem + (size_t)(wib * NSLOT) * ROW_BYTES;

    int  idxv[NSLOT];
    bool validv[NSLOT];

    // phase A: issue gather loads into LDS
    for (int j = 0; j < NSLOT; ++j) {
        int pix = pix0 + j;
        int idx = IGNORE_LABEL;
        if (pix < P) idx = __builtin_amdgcn_readfirstlane(idx_img[pix]);
        bool valid = (pix < P) && (idx != IGNORE_LABEL);
        idxv[j] = idx;
        validv[j] = valid;
        if (valid) {
            const float* src = feats + (size_t)idx * D + lane * 4;   // 16B/lane
            void* dst = slotbase + j * ROW_BYTES + lane * 16;
            __builtin_amdgcn_global_load_async_to_lds_b128(
                as_gv4(src), as_lv4(dst), /*offset=*/0, /*cpol=*/0);
        }
    }
    wait_async0();   // LDS rows are now valid

    // phase B: stream rows to the output
    for (int j = 0; j < NSLOT; ++j) {
        int pix = pix0 + j;
        if (pix >= P) break;
        float* dstrow = proj + (size_t)pix * D + lane * 4;
        if (validv[j]) {
            void* src = slotbase + j * ROW_BYTES + lane * 16;
            __builtin_amdgcn_global_store_async_from_lds_b128(
                as_gv4(dstrow), as_lv4(src), /*offset=*/0, /*cpol=TH_NT*/1);
        } else {
            __builtin_nontemporal_store((f4)0.0f, (f4*)dstrow);
        }
        if (lane == 0) {
            __builtin_nontemporal_store((float)idxv[j], idx_out + pix);
            if (validv[j]) atomicAdd(counts + idxv[j], 1);
        }
    }
    wait_async0();   // drain before wave exit
}
#else
// ---------------------------------------------------------------------------
// Fallback: classic VGPR path. One wave32 per pixel,
// float4 per lane, NT stores keep the 51MB feature table resident in L2.
// ---------------------------------------------------------------------------
__global__ __launch_bounds__(256) void RaycastGather_kernel(
    const float* __restrict__ feats,
    const int*   __restrict__ idx_img,
    float*       __restrict__ proj,
    float*       __restrict__ idx_out,
    int*         __restrict__ counts,
    int P) {
    int gtid = blockIdx.x * blockDim.x + threadIdx.x;
    int pix  = gtid >> 5;
    int lane = gtid & 31;
    if (pix >= P) return;

    int idx = __builtin_amdgcn_readfirstlane(idx_img[pix]);
    bool valid = (idx != IGNORE_LABEL);

    f4 v = (f4)0.0f;
    if (valid) {
        const f4* row = (const f4*)(feats + (size_t)idx * D);
        v = row[lane];
    }
    f4* outp = (f4*)(proj + (size_t)pix * D);
    __builtin_nontemporal_store(v, outp + lane);

    if (lane == 0) {
        __builtin_nontemporal_store((float)idx, idx_out + pix);
        if (valid) atomicAdd(counts + idx, 1);
    }
}
#endif

// ---------------------------------------------------------------------------
// Kernel 2: int bins -> float in place (exact for counts << 2^24).
// ---------------------------------------------------------------------------
__global__ __launch_bounds__(256) void RaycastCountsToFloat_kernel(float* __restrict__ counts,
                                                                   int V) {
    int i = blockIdx.x * blockDim.x + threadIdx.x;
    if (i < V) {
        int c = ((const int*)counts)[i];
        counts[i] = (float)c;
    }
}

extern "C" void kernel_launch(void* const* d_in, const int* in_sizes, int n_in,
                              void* d_out, int out_size, void* d_ws, size_t ws_size,
                              hipStream_t stream) {
    const float* feats   = (const float*)d_in[0];  // [V*128] f32
    const int*   idx_img = (const int*)d_in[1];    // [P] i32

    const int V = in_sizes[0] / D;                 // 100000
    const int P = in_sizes[1];                     // 614400

    // d_out layout (all float32): [P*D projected][P indexes][V counts]
    float* proj    = (float*)d_out;
    float* idx_out = proj + (size_t)P * D;
    float* cnt_f   = idx_out + P;
    int*   cnt_i   = (int*)cnt_f;

    {   // 0) zero histogram bins
        int threads = 256;
        int blocks  = (V + threads - 1) / threads;
        RaycastZeroCounts_kernel<<<blocks, threads, 0, stream>>>(cnt_i, V);
    }
#if defined(HAVE_GFX1250_ASYNC)
    {   // 1) async-engine gather: NSLOT pixels per wave
        int pix_per_block = NSLOT * WAVES_PER_BLOCK;               // 32
        int blocks = (P + pix_per_block - 1) / pix_per_block;      // 19200
        RaycastGatherAsync_kernel<<<blocks, 256, 0, stream>>>(
            feats, idx_img, proj, idx_out, cnt_i, P);
    }
#else
    {   // 1) fallback VGPR gather: one wave per pixel
        long long total = (long long)P * 32;
        int threads = 256;
        int blocks  = (int)((total + threads - 1) / threads);
        RaycastGather_kernel<<<blocks, threads, 0, stream>>>(
            feats, idx_img, proj, idx_out, cnt_i, P);
    }
#endif
    {   // 2) int counts -> float, in place
        int threads = 256;
        int blocks  = (V + threads - 1) / threads;
        RaycastCountsToFloat_kernel<<<blocks, threads, 0, stream>>>(cnt_f, V);
    }
}